// CWT_2826088480932
// MI455X (gfx1250) — compile-verified
//
#include <hip/hip_runtime.h>

typedef __attribute__((ext_vector_type(2))) float v2f;
typedef __attribute__((ext_vector_type(8))) float v8f;

#define SIG_LEN 2048
#define UP      4096
#define N1PAD   1024
#define NA_SC   256
#define NBATCH  32

// workspace layout (float offsets)
#define WS_FRE   0              // DFT16 matrix, cos, 256
#define WS_FIM   256            // DFT16 matrix, sin (inverse sign +), 256
#define WS_TW0RE 512            // stage0 twiddles cos, 4096
#define WS_TW0IM 4608           // stage0 twiddles sin, 4096
#define WS_TW1RE 8704           // stage1 twiddles cos, 256
#define WS_TW1IM 8960           // stage1 twiddles sin, 256
#define WS_XHRE  9216           // spectrum real, 32*4096
#define WS_XHIM  (9216 + NBATCH * UP)

// ---------------------------------------------------------------------------
// Precompute DFT-16 matrix and Stockham twiddle tables (exact, via double).
// ---------------------------------------------------------------------------
__global__ void cwt_init_tables(float* __restrict__ ws) {
    const double PI2 = 6.283185307179586476925286766559;
    const int t = blockIdx.x * blockDim.x + threadIdx.x;
    const int stride = gridDim.x * blockDim.x;
    for (int i = t; i < 256; i += stride) {
        int u = i >> 4, c = i & 15;
        double ang = PI2 * (double)(u * c) / 16.0;
        ws[WS_FRE + i] = (float)cos(ang);
        ws[WS_FIM + i] = (float)sin(ang);
    }
    for (int i = t; i < 4096; i += stride) {
        int u = i >> 8, p = i & 255;
        double ang = PI2 * (double)(u * p) / 4096.0;
        ws[WS_TW0RE + i] = (float)cos(ang);
        ws[WS_TW0IM + i] = (float)sin(ang);
    }
    for (int i = t; i < 256; i += stride) {
        int u = i >> 4, p = i & 15;
        double ang = PI2 * (double)(u * p) / 256.0;
        ws[WS_TW1RE + i] = (float)cos(ang);
        ws[WS_TW1IM + i] = (float)sin(ang);
    }
}

// ---------------------------------------------------------------------------
// Load DFT-16 matrix into WMMA A-operand registers (16x4 f32 chunks).
// A layout (ISA 7.12.2): lanes 0-15 -> M=0-15, VGPR0 = K{0 | 2}, VGPR1 = K{1 | 3}
// sgn = -1 for forward FFT (conjugate), +1 for inverse.
// ---------------------------------------------------------------------------
__device__ __forceinline__ void load_dft16(const float* __restrict__ ws, float sgn,
                                           int lane, v2f aRe[4], v2f aIm[4], v2f aSum[4]) {
    const int row  = lane & 15;
    const int half = lane >> 4;
#pragma unroll
    for (int k = 0; k < 4; ++k) {
        const int c0 = 4 * k + 2 * half;
        aRe[k].x = ws[WS_FRE + row * 16 + c0];
        aRe[k].y = ws[WS_FRE + row * 16 + c0 + 1];
        aIm[k].x = sgn * ws[WS_FIM + row * 16 + c0];
        aIm[k].y = sgn * ws[WS_FIM + row * 16 + c0 + 1];
        aSum[k]  = aRe[k] + aIm[k];
    }
}

// ---------------------------------------------------------------------------
// One Stockham radix-16 stage for N=4096 (= 16^3), computed as a complex
// 16x16 (DFT matrix) x 16x256 (data) matmul on V_WMMA_F32_16X16X4_F32,
// followed by twiddle multiply + permuted store.
//   Stage 0: n=4096, s=1   -> y[16p+u]        * w4096^{up}
//   Stage 1: n=256,  s=16  -> y[q+16u+256p]   * w256^{up}
//   Stage 2: n=16,   s=256 -> y[j+256u]       (no twiddle, natural order)
// All stages read M[c][j] = src[j + 256 c].
// Complex product via 3 real WMMAs (Karatsuba): re=P-M, im=S-P-M.
// ---------------------------------------------------------------------------
template <int STAGE>
__device__ __forceinline__ void radix16_stage(
    const float* __restrict__ sre, const float* __restrict__ sim,
    float* __restrict__ dre, float* __restrict__ dim_,
    const v2f aRe[4], const v2f aIm[4], const v2f aSum[4],
    const float* __restrict__ twre, const float* __restrict__ twim,
    float tsign, int lane, int wave) {
    const int col  = lane & 15;
    const int half = lane >> 4;
#pragma unroll
    for (int i = 0; i < 2; ++i) {
        const int tile = wave + 8 * i;       // 8 waves x 2 tiles = 16 column tiles
        const int j = tile * 16 + col;
        v8f accP = {}, accM = {}, accS = {};
#pragma unroll
        for (int k = 0; k < 4; ++k) {
            // B layout (4x16 f32): VGPR0 = row {4k | 4k+2}, VGPR1 = row {4k+1 | 4k+3}
            const int kb  = 4 * k + 2 * half;
            const int idx = j + 256 * kb;
            v2f br, bi;
            br.x = sre[idx]; br.y = sre[idx + 256];
            bi.x = sim[idx]; bi.y = sim[idx + 256];
            const v2f bs = br + bi;
            accP = __builtin_amdgcn_wmma_f32_16x16x4_f32(false, aRe[k],  false, br,
                                                         (short)0, accP, false, false);
            accM = __builtin_amdgcn_wmma_f32_16x16x4_f32(false, aIm[k],  false, bi,
                                                         (short)0, accM, false, false);
            accS = __builtin_amdgcn_wmma_f32_16x16x4_f32(false, aSum[k], false, bs,
                                                         (short)0, accS, false, false);
        }
        // C/D layout: lane holds col = lane&15, VGPR v holds row u = v + 8*(lane>>4)
#pragma unroll
        for (int v = 0; v < 8; ++v) {
            const int u = v + 8 * half;
            const float cr = accP[v] - accM[v];
            const float ci = accS[v] - accP[v] - accM[v];
            float yr, yi;
            int o;
            if (STAGE == 0) {
                const int p = j;
                const float twr = twre[u * 256 + p];
                const float twi = tsign * twim[u * 256 + p];
                yr = cr * twr - ci * twi;
                yi = cr * twi + ci * twr;
                o = 16 * p + u;
            } else if (STAGE == 1) {
                const int p = tile;          // j = 16*tile + col -> p = tile, q = col
                const float twr = twre[u * 16 + p];
                const float twi = tsign * twim[u * 16 + p];
                yr = cr * twr - ci * twi;
                yi = cr * twi + ci * twr;
                o = col + 16 * u + 256 * p;
            } else {
                yr = cr; yi = ci;
                o = j + 256 * u;
            }
            dre[o]  = yr;
            dim_[o] = yi;
        }
    }
}

// ---------------------------------------------------------------------------
// Forward FFT of the 32 reflect-padded signals -> spectrum planes in ws.
// ---------------------------------------------------------------------------
__global__ __launch_bounds__(256) void cwt_forward_fft(const float* __restrict__ x,
                                                       float* __restrict__ ws) {
    __shared__ float lds[4 * UP];            // 64 KB: ping re/im + pong re/im
    float* pr = lds;
    float* pi = lds + UP;
    float* qr = lds + 2 * UP;
    float* qi = lds + 3 * UP;

    const int b    = blockIdx.x;
    const int lane = threadIdx.x & 31;
    const int wave = threadIdx.x >> 5;

    v2f aRe[4], aIm[4], aSum[4];
    load_dft16(ws, -1.0f, lane, aRe, aIm, aSum);   // forward: conjugated DFT matrix

    for (int j = threadIdx.x; j < UP; j += 256) {  // reflect pad
        int m = j - N1PAD;
        m = (m < 0) ? -m : m;
        m = (m >= SIG_LEN) ? (2 * SIG_LEN - 2 - m) : m;
        pr[j] = x[b * SIG_LEN + m];
        pi[j] = 0.0f;
    }
    __syncthreads();
    radix16_stage<0>(pr, pi, qr, qi, aRe, aIm, aSum, ws + WS_TW0RE, ws + WS_TW0IM, -1.0f, lane, wave);
    __syncthreads();
    radix16_stage<1>(qr, qi, pr, pi, aRe, aIm, aSum, ws + WS_TW1RE, ws + WS_TW1IM, -1.0f, lane, wave);
    __syncthreads();
    radix16_stage<2>(pr, pi, qr, qi, aRe, aIm, aSum, nullptr, nullptr, -1.0f, lane, wave);
    __syncthreads();
    for (int k = threadIdx.x; k < UP; k += 256) {
        ws[WS_XHRE + b * UP + k] = qr[k];
        ws[WS_XHIM + b * UP + k] = qi[k];
    }
}

// ---------------------------------------------------------------------------
// Main kernel: per (scale a, batch b) multiply spectrum by Psih[a] (and 1/N),
// inverse FFT via WMMA radix-16, write middle 2048 complex samples.
// ---------------------------------------------------------------------------
__global__ __launch_bounds__(256) void cwt_ifft_scales(const float* __restrict__ Psih,
                                                       const float* __restrict__ ws,
                                                       float* __restrict__ out) {
    __shared__ float lds[4 * UP];
    float* pr = lds;
    float* pi = lds + UP;
    float* qr = lds + 2 * UP;
    float* qi = lds + 3 * UP;

    const int a    = blockIdx.x;
    const int b    = blockIdx.y;
    const int lane = threadIdx.x & 31;
    const int wave = threadIdx.x >> 5;

    v2f aRe[4], aIm[4], aSum[4];
    load_dft16(ws, 1.0f, lane, aRe, aIm, aSum);    // inverse: +i sign

    const float invN = 1.0f / (float)UP;           // jnp.fft.ifft 1/N folded in here
    const float* __restrict__ psi = Psih + (size_t)a * UP;
    const float* __restrict__ xr  = ws + WS_XHRE + (size_t)b * UP;
    const float* __restrict__ xi  = ws + WS_XHIM + (size_t)b * UP;
    for (int k = threadIdx.x; k < UP; k += 256) {
        const float p = psi[k] * invN;
        pr[k] = p * xr[k];
        pi[k] = p * xi[k];
    }
    __syncthreads();
    radix16_stage<0>(pr, pi, qr, qi, aRe, aIm, aSum, ws + WS_TW0RE, ws + WS_TW0IM, 1.0f, lane, wave);
    __syncthreads();
    radix16_stage<1>(qr, qi, pr, pi, aRe, aIm, aSum, ws + WS_TW1RE, ws + WS_TW1IM, 1.0f, lane, wave);
    __syncthreads();
    radix16_stage<2>(pr, pi, qr, qi, aRe, aIm, aSum, nullptr, nullptr, 1.0f, lane, wave);
    __syncthreads();

    // output (B, NA, L) complex64 -> interleaved float2, coalesced
    float2* __restrict__ o = (float2*)out + ((size_t)b * NA_SC + a) * SIG_LEN;
    for (int n = threadIdx.x; n < SIG_LEN; n += 256) {
        float2 v;
        v.x = qr[N1PAD + n];
        v.y = qi[N1PAD + n];
        o[n] = v;
    }
}

// ---------------------------------------------------------------------------
extern "C" void kernel_launch(void* const* d_in, const int* in_sizes, int n_in,
                              void* d_out, int out_size, void* d_ws, size_t ws_size,
                              hipStream_t stream) {
    (void)in_sizes; (void)n_in; (void)out_size; (void)ws_size;
    const float* x    = (const float*)d_in[0];   // (32, 2048) f32
    const float* Psih = (const float*)d_in[1];   // (256, 4096) f32
    float*       out  = (float*)d_out;           // (32, 256, 2048) complex64 interleaved
    float*       ws   = (float*)d_ws;

    hipLaunchKernelGGL(cwt_init_tables, dim3(16), dim3(256), 0, stream, ws);
    hipLaunchKernelGGL(cwt_forward_fft, dim3(NBATCH), dim3(256), 0, stream, x, ws);
    hipLaunchKernelGGL(cwt_ifft_scales, dim3(NA_SC, NBATCH), dim3(256), 0, stream,
                       Psih, ws, out);
}